// KSLinearTriton_35716948033572
// MI455X (gfx1250) — compile-verified
//
#include <hip/hip_runtime.h>
#include <hip/hip_bf16.h>

typedef float v2f __attribute__((ext_vector_type(2)));
typedef float v4f __attribute__((ext_vector_type(4)));
typedef float v8f __attribute__((ext_vector_type(8)));

#define MTILE 16
#define FS    2052              // 2048 + 4 pad (== 4 mod 64 -> conflict-free layer0)
#define LDS_BYTES (MTILE * FS * 4)

__device__ __forceinline__ v8f wmma4(v2f a, v2f b, v8f c) {
  // (neg_a, A, neg_b, B, c_mod, C, reuse_a, reuse_b)
  return __builtin_amdgcn_wmma_f32_16x16x4_f32(false, a, false, b, (short)0, c,
                                               false, false);
}

__global__ __launch_bounds__(256) void ks_fused_kernel(
    const float* __restrict__ x, const float* __restrict__ w0,
    const float* __restrict__ w1, const float* __restrict__ bias,
    float* __restrict__ out) {
  extern __shared__ float smem[];  // [MTILE][FS], feature stored as f' = d*32 + c

  const int t    = threadIdx.x;
  const int row0 = blockIdx.x * MTILE;

  // ---------------- global -> LDS copy (transposing f=c*64+d -> f'=d*32+c) ----
  // lane group: c = t&31 (LDS banks = c -> conflict free), g = t>>5 owns rows g,g+8
  // x is a 256MB single-use stream: non-temporal loads keep L2 for the weights.
  {
    const int c = t & 31;
    const int g = t >> 5;
#pragma unroll
    for (int rm = 0; rm < 2; ++rm) {
      const int m = g + rm * 8;
      const v4f* src =
          reinterpret_cast<const v4f*>(x + (size_t)(row0 + m) * 2048 + c * 64);
      float* dst = smem + m * FS + c;
#pragma unroll
      for (int d4 = 0; d4 < 16; ++d4) {
        v4f v = __builtin_nontemporal_load(&src[d4]);
        dst[(d4 * 4 + 0) * 32] = v.x;
        dst[(d4 * 4 + 1) * 32] = v.y;
        dst[(d4 * 4 + 2) * 32] = v.z;
        dst[(d4 * 4 + 3) * 32] = v.w;
      }
    }
  }
  __syncthreads();

  const int lane = t & 31;
  const int wv   = t >> 5;   // wave id 0..7
  const int half = lane >> 4;
  const int lh   = lane & 15;

  // ---------------- layer 0: per d, Y[m,b,d] = sum_c X[m,c,d] * w0[d,c,b] ----
  // in-place in LDS: slot f' = d*32 + b replaces f' = d*32 + c (disjoint d per wave)
  const float* Arow = smem + lh * FS;
#pragma unroll 2
  for (int dd = 0; dd < 8; ++dd) {
    const int d = wv * 8 + dd;
    const float* Ap = Arow + d * 32;      // + c (contiguous K!)
    const float* Bp = w0 + d * 1024;      // + c*32 + b
    v8f acc0 = {};
    v8f acc1 = {};
#pragma unroll
    for (int k = 0; k < 32; k += 4) {
      const int kk = k + 2 * half;        // lanes 0-15: K=k,k+1 ; 16-31: K=k+2,k+3
      v2f A  = {Ap[kk], Ap[kk + 1]};
      v2f B0 = {Bp[kk * 32 + lh],      Bp[(kk + 1) * 32 + lh]};
      v2f B1 = {Bp[kk * 32 + 16 + lh], Bp[(kk + 1) * 32 + 16 + lh]};
      acc0 = wmma4(A, B0, acc0);
      acc1 = wmma4(A, B1, acc1);
    }
    float* Yp = smem + d * 32;            // D: lane<16 -> (M=v,N=lh), else (v+8,lh)
#pragma unroll
    for (int v = 0; v < 8; ++v) {
      Yp[(v + 8 * half) * FS + lh]      = acc0[v];
      Yp[(v + 8 * half) * FS + 16 + lh] = acc1[v];
    }
  }
  __syncthreads();

  // ---------------- layer 1: per a, Out[m,a*32+b] = sum_c Y[m,a*32+c]*w1[a,c,b]+bias
  // input feature f = a*32+c lives at f' = 1024*(a&1) + c*32 + (a>>1)
#pragma unroll 2
  for (int aa = 0; aa < 8; ++aa) {
    const int a = wv * 8 + aa;
    const float* Ap = Arow + 1024 * (a & 1) + (a >> 1);  // + c*32
    const float* Bp = w1 + a * 1024;
    const float bs0 = bias[a * 32 + lh];
    const float bs1 = bias[a * 32 + 16 + lh];
    v8f acc0, acc1;
#pragma unroll
    for (int v = 0; v < 8; ++v) { acc0[v] = bs0; acc1[v] = bs1; }
#pragma unroll
    for (int k = 0; k < 32; k += 4) {
      const int kk = k + 2 * half;
      v2f A  = {Ap[kk * 32], Ap[(kk + 1) * 32]};
      v2f B0 = {Bp[kk * 32 + lh],      Bp[(kk + 1) * 32 + lh]};
      v2f B1 = {Bp[kk * 32 + 16 + lh], Bp[(kk + 1) * 32 + 16 + lh]};
      acc0 = wmma4(A, B0, acc0);
      acc1 = wmma4(A, B1, acc1);
    }
    float* Op = out + (size_t)row0 * 2048 + a * 32;
#pragma unroll
    for (int v = 0; v < 8; ++v) {
      __builtin_nontemporal_store(acc0[v], &Op[(size_t)(v + 8 * half) * 2048 + lh]);
      __builtin_nontemporal_store(acc1[v], &Op[(size_t)(v + 8 * half) * 2048 + 16 + lh]);
    }
  }
}

extern "C" void kernel_launch(void* const* d_in, const int* in_sizes, int n_in,
                              void* d_out, int out_size, void* d_ws, size_t ws_size,
                              hipStream_t stream) {
  const float* x    = (const float*)d_in[0];
  const float* w0   = (const float*)d_in[1];
  const float* w1   = (const float*)d_in[2];
  const float* bias = (const float*)d_in[3];
  float* out        = (float*)d_out;

  const int batch = in_sizes[0] / 2048;        // 32768
  const int grid  = batch / MTILE;             // 2048 workgroups of 8 waves

  (void)hipFuncSetAttribute(reinterpret_cast<const void*>(ks_fused_kernel),
                            hipFuncAttributeMaxDynamicSharedMemorySize, LDS_BYTES);

  ks_fused_kernel<<<grid, 256, LDS_BYTES, stream>>>(x, w0, w1, bias, out);
}